// GATEncoder_59760174957312
// MI455X (gfx1250) — compile-verified
//
#include <hip/hip_runtime.h>
#include <hip/hip_bf16.h>
#include <stdint.h>

// ---------------- problem constants (from reference) ----------------
#define Bsz     16
#define Nn      768
#define IN_DIM  16
#define HID     64
#define EMB     128
#define HEADS   8
#define F1      (HID * HEADS)   // 512
#define MROWS   (Bsz * Nn)      // 12288

typedef float v2f __attribute__((ext_vector_type(2)));
typedef float v8f __attribute__((ext_vector_type(8)));
typedef unsigned int v4u  __attribute__((ext_vector_type(4)));
typedef int          v8i_t __attribute__((ext_vector_type(8)));
typedef int          v4i_t __attribute__((ext_vector_type(4)));

// D = A(16x4,f32) * B(4x16,f32) + C(16x16,f32)
__device__ __forceinline__ v8f wmma_f32_k4(v2f a, v2f b, v8f c) {
  return __builtin_amdgcn_wmma_f32_16x16x4_f32(
      /*neg_a=*/false, a, /*neg_b=*/false, b,
      /*c_mod=*/(short)0, c, /*reuse_a=*/false, /*reuse_b=*/false);
}

// =====================================================================
// K1: h0[m, o] = x[m, :16] @ ip_w[16, 64] + ip_b[o]
// grid(768, 4), block(32): one wave per 16x16 tile, K=16 in 4 steps
// =====================================================================
__global__ void k_in_proj(const float* __restrict__ x,
                          const float* __restrict__ w,
                          const float* __restrict__ bias,
                          float* __restrict__ h0) {
  const int m0   = blockIdx.x * 16;
  const int n0   = blockIdx.y * 16;
  const int lane = threadIdx.x & 31;
  const int r    = lane & 15;
  const int kb   = (lane >> 4) * 2;
  const int col  = r;
  v8f acc = {};
#pragma unroll
  for (int k0 = 0; k0 < IN_DIM; k0 += 4) {
    v2f a, b;
    a.x = x[(m0 + r) * IN_DIM + k0 + kb];
    a.y = x[(m0 + r) * IN_DIM + k0 + kb + 1];
    b.x = w[(k0 + kb) * HID + n0 + col];
    b.y = w[(k0 + kb + 1) * HID + n0 + col];
    acc = wmma_f32_k4(a, b, acc);
  }
  const float bv    = bias[n0 + col];
  const int   rbase = (lane >> 4) * 8;
#pragma unroll
  for (int v = 0; v < 8; ++v)
    h0[(size_t)(m0 + rbase + v) * HID + n0 + col] = acc[v] + bv;
}

// =====================================================================
// K2: Wh[b,h,n,o] = hin[b,n,:F] @ W[h,:F,:64]
// grid(MROWS/16, HEADS), block(128): wave w covers o-tile w
// =====================================================================
__global__ void k_wh(const float* __restrict__ hin,
                     const float* __restrict__ W,
                     float* __restrict__ Wh, int F) {
  const int m0   = blockIdx.x * 16;
  const int head = blockIdx.y;
  const int wave = threadIdx.x >> 5;
  const int lane = threadIdx.x & 31;
  const int n0   = wave * 16;
  const int r    = lane & 15;
  const int kb   = (lane >> 4) * 2;
  const int col  = r;
  const float* __restrict__ Wt = W + (size_t)head * F * HID;
  v8f acc = {};
  for (int k0 = 0; k0 < F; k0 += 4) {
    __builtin_prefetch(&hin[(size_t)(m0 + r) * F + k0 + 16], 0, 0);
    v2f a, b;
    a.x = hin[(size_t)(m0 + r) * F + k0 + kb];
    a.y = hin[(size_t)(m0 + r) * F + k0 + kb + 1];
    b.x = Wt[(size_t)(k0 + kb) * HID + n0 + col];
    b.y = Wt[(size_t)(k0 + kb + 1) * HID + n0 + col];
    acc = wmma_f32_k4(a, b, acc);
  }
  const int b_idx = m0 / Nn;
  const int nrow  = m0 % Nn;
  const int rbase = (lane >> 4) * 8;
  float* __restrict__ out =
      Wh + (((size_t)b_idx * HEADS + head) * Nn + nrow) * HID;
#pragma unroll
  for (int v = 0; v < 8; ++v)
    out[(size_t)(rbase + v) * HID + n0 + col] = acc[v];
}

// =====================================================================
// K3: fused GAT attention for one (b, head):
//   s1/s2 = Wh @ a1/a2 ; P = exp(leakyrelu(s1[n]+s2[m])) built in regs;
//   hp = (P @ Wh) / rowsum(P).  P@Wh via 4 f32 WMMAs per K-step; the
//   softmax denominator Z is accumulated in VALU from the A-fragment
//   (zpart += a.x + a.y) and reduced with wave shuffles — saves 1 of 5
//   WMMAs on the low-rate f32 matrix pipe.
// grid(B*HEADS), block(256), dyn LDS = 768*64*4 + 2*768*4 bytes
// Wh[b,h] is DMA'd into LDS with the Tensor Data Mover.
// =====================================================================
__global__ __launch_bounds__(256) void k_att(const float* __restrict__ Wh,
                                             const float* __restrict__ a_vec,
                                             float* __restrict__ hout) {
  extern __shared__ float smem[];
  float* Wh_s = smem;                 // Nn*HID floats, at LDS offset 0
  float* s1_s = smem + Nn * HID;      // Nn floats
  float* s2_s = s1_s + Nn;            // Nn floats

  const int bh   = blockIdx.x;
  const int b    = bh >> 3;
  const int head = bh & 7;
  const float* __restrict__ Whg = Wh + (size_t)bh * Nn * HID;

  // ---- TDM: DMA 768x64 f32 tile (192KB) global -> LDS, wave 0 only ----
  if (threadIdx.x < 32) {
    const unsigned long long ga = (unsigned long long)(uintptr_t)Whg;
    const unsigned nelem = Nn * HID;              // 49152 elements, 4B each
    v4u g0;
    g0[0] = 1u;                                    // count=1, user descriptor
    g0[1] = 0u;                                    // lds_addr = dyn-LDS base
    g0[2] = (unsigned)(ga & 0xFFFFFFFFull);        // global_addr[31:0]
    g0[3] = (unsigned)((ga >> 32) & 0x01FFFFFFull) // global_addr[56:32]
            | (2u << 30);                          // type = 2 ("image")
    v8i_t g1;
    g1[0] = (int)(2u << 16);                       // wg_mask=0, data_size=4B
    g1[1] = (int)((nelem & 0xFFFFu) << 16);        // tensor_dim0[15:0]
    g1[2] = (int)(((nelem >> 16) & 0xFFFFu)        // tensor_dim0[31:16]
                  | (1u << 16));                   // tensor_dim1 = 1
    g1[3] = (int)((nelem & 0xFFFFu) << 16);        // tile_dim0 = nelem
    g1[4] = 1;                                     // tile_dim1 = 1
    g1[5] = (int)nelem;                            // tensor_dim0_stride lo
    g1[6] = 0;                                     // stride hi / dim1 stride
    g1[7] = 0;
    v4i_t gz4 = {0, 0, 0, 0};
    v8i_t gz8 = {0, 0, 0, 0, 0, 0, 0, 0};
    __builtin_amdgcn_tensor_load_to_lds(g0, g1, gz4, gz4, gz8, 0);
    __builtin_amdgcn_s_wait_tensorcnt(0);
  }
  __syncthreads();

  // ---- score vectors: s1[n] = Wh[n,:] . a1 ; s2[n] = Wh[n,:] . a2 ----
  const float* __restrict__ av = a_vec + head * (2 * HID);
  for (int n = threadIdx.x; n < Nn; n += 256) {
    float d1 = 0.f, d2 = 0.f;
#pragma unroll 8
    for (int f = 0; f < HID; ++f) {
      const float wv = Wh_s[n * HID + f];
      d1 += wv * av[f];
      d2 += wv * av[HID + f];
    }
    s1_s[n] = d1;
    s2_s[n] = d2;
  }
  __syncthreads();

  // ---- main flash-style loop: each wave owns 16-row output tiles ----
  const int wave  = threadIdx.x >> 5;
  const int lane  = threadIdx.x & 31;
  const int r     = lane & 15;
  const int hi    = lane >> 4;
  const int kb    = hi * 2;
  const int col   = r;
  const int rbase = hi * 8;

  for (int mt = wave; mt < Nn / 16; mt += 8) {
    const int   n0  = mt * 16;
    const float s1l = s1_s[n0 + r];
    v8f acc0 = {}, acc1 = {}, acc2 = {}, acc3 = {};
    float zpart = 0.f;   // partial row-sum of P for row (lane&15)
    for (int m = 0; m < Nn; m += 4) {
      const int mA = m + kb;
      float e0 = s1l + s2_s[mA];
      float e1 = s1l + s2_s[mA + 1];
      e0 = (e0 > 0.f) ? e0 : 0.2f * e0;           // leaky_relu(0.2)
      e1 = (e1 > 0.f) ? e1 : 0.2f * e1;
      v2f a;
      a.x = __expf(e0);
      a.y = __expf(e1);
      zpart += a.x + a.y;
      const float* __restrict__ w0 = &Wh_s[mA * HID + col];
      const float* __restrict__ w1 = &Wh_s[(mA + 1) * HID + col];
      v2f b0, b1, b2, b3;
      b0.x = w0[0];  b0.y = w1[0];
      b1.x = w0[16]; b1.y = w1[16];
      b2.x = w0[32]; b2.y = w1[32];
      b3.x = w0[48]; b3.y = w1[48];
      acc0 = wmma_f32_k4(a, b0, acc0);
      acc1 = wmma_f32_k4(a, b1, acc1);
      acc2 = wmma_f32_k4(a, b2, acc2);
      acc3 = wmma_f32_k4(a, b3, acc3);
    }
    // combine the two half-wave partials: lane L and L^16 hold the same row
    const float zfull = zpart + __shfl_xor(zpart, 16, 32);
    // write: hout[(b*Nn + n0 + row) * F1 + head*HID + {0,16,32,48} + col]
    float* __restrict__ op =
        hout + ((size_t)(b * Nn + n0 + rbase)) * F1 + head * HID + col;
#pragma unroll
    for (int v = 0; v < 8; ++v) {
      // Z for C-layout row (rbase + v) lives in lane (rbase + v)
      const float zv  = __shfl(zfull, rbase + v, 32);
      const float inv = 1.f / zv;
      op[(size_t)v * F1 + 0]  = acc0[v] * inv;
      op[(size_t)v * F1 + 16] = acc1[v] * inv;
      op[(size_t)v * F1 + 32] = acc2[v] * inv;
      op[(size_t)v * F1 + 48] = acc3[v] * inv;
    }
  }
}

// =====================================================================
// K4: out[m, e] = hcat[m, :512] @ proj_w[512, 128] + proj_b[e]
// grid(MROWS/16, 2), block(128): wave w -> e-tile (blockIdx.y*4 + w)
// =====================================================================
__global__ void k_out_proj(const float* __restrict__ hcat,
                           const float* __restrict__ w,
                           const float* __restrict__ bias,
                           float* __restrict__ out) {
  const int m0   = blockIdx.x * 16;
  const int wave = threadIdx.x >> 5;
  const int lane = threadIdx.x & 31;
  const int n0   = (blockIdx.y * 4 + wave) * 16;
  const int r    = lane & 15;
  const int kb   = (lane >> 4) * 2;
  const int col  = r;
  v8f acc = {};
  for (int k0 = 0; k0 < F1; k0 += 4) {
    __builtin_prefetch(&hcat[(size_t)(m0 + r) * F1 + k0 + 16], 0, 0);
    v2f a, b;
    a.x = hcat[(size_t)(m0 + r) * F1 + k0 + kb];
    a.y = hcat[(size_t)(m0 + r) * F1 + k0 + kb + 1];
    b.x = w[(size_t)(k0 + kb) * EMB + n0 + col];
    b.y = w[(size_t)(k0 + kb + 1) * EMB + n0 + col];
    acc = wmma_f32_k4(a, b, acc);
  }
  const float bv    = bias[n0 + col];
  const int   rbase = (lane >> 4) * 8;
#pragma unroll
  for (int v = 0; v < 8; ++v)
    out[(size_t)(m0 + rbase + v) * EMB + n0 + col] = acc[v] + bv;
}

// =====================================================================
extern "C" void kernel_launch(void* const* d_in, const int* in_sizes, int n_in,
                              void* d_out, int out_size, void* d_ws, size_t ws_size,
                              hipStream_t stream) {
  const float* x      = (const float*)d_in[0];
  const float* ip_w   = (const float*)d_in[1];
  const float* ip_b   = (const float*)d_in[2];
  const float* gat_w0 = (const float*)d_in[3];
  const float* gat_a0 = (const float*)d_in[4];
  const float* gat_w1 = (const float*)d_in[5];
  const float* gat_a1 = (const float*)d_in[6];
  const float* proj_w = (const float*)d_in[7];
  const float* proj_b = (const float*)d_in[8];
  float* out = (float*)d_out;

  // workspace layout
  char*  ws   = (char*)d_ws;
  float* h0   = (float*)(ws);                                   // 3 MB
  float* Wh   = (float*)(ws + (size_t)MROWS * HID * 4);         // 25.2 MB
  float* hcat = (float*)(ws + (size_t)MROWS * HID * 4
                            + (size_t)MROWS * HEADS * HID * 4); // 25.2 MB

  const size_t att_lds = ((size_t)Nn * HID + 2 * Nn) * sizeof(float);

  // 1) input projection
  k_in_proj<<<dim3(MROWS / 16, HID / 16), 32, 0, stream>>>(x, ip_w, ip_b, h0);
  // 2) layer 0: Wh then fused attention
  k_wh<<<dim3(MROWS / 16, HEADS), 128, 0, stream>>>(h0, gat_w0, Wh, HID);
  k_att<<<dim3(Bsz * HEADS), 256, att_lds, stream>>>(Wh, gat_a0, hcat);
  // 3) layer 1: Wh (F=512) then fused attention (reuse hcat as output)
  k_wh<<<dim3(MROWS / 16, HEADS), 128, 0, stream>>>(hcat, gat_w1, Wh, F1);
  k_att<<<dim3(Bsz * HEADS), 256, att_lds, stream>>>(Wh, gat_a1, hcat);
  // 4) output projection
  k_out_proj<<<dim3(MROWS / 16, EMB / 64), 128, 0, stream>>>(hcat, proj_w,
                                                             proj_b, out);
}